// CenterLoss_8040178778749
// MI455X (gfx1250) — compile-verified
//
#include <hip/hip_runtime.h>

// CenterLoss, algebraically reduced:
//   loss = ( sum_i clip(||x_i - c_{labels[i]}||^2, 1e-12, 1e12) + B*(C-1)*1e-12 ) / B
// Traffic: ~1MB (x + 1024 gathered center rows) instead of >=410MB for the
// naive [B,C] distmat -> launch-latency bound on MI455X (23.3 TB/s => ~45ns).
// Row-wise sum-of-squares is computed as the diagonal of E*E^T via
// V_WMMA_F32_16X16X4_F32 (K accumulated 128 in steps of 4). For a Gram
// product the A and B wmma fragments are the identical register pair.

#define NUM_CLASSES 100000
#define FEAT_DIM    128
#define BATCH       1024

#define ROWS_PER_WAVE   16
#define WAVES_PER_BLOCK 4          // 128 threads/block, 4 waves
#define NUM_WAVES       (BATCH / ROWS_PER_WAVE)   // 64 partials
#define LDS_STRIDE      132        // 128 + 4 pad: ds reads of col c hit bank (c + 4*row)%64 -> conflict-free

typedef __attribute__((ext_vector_type(2))) float v2f;
typedef __attribute__((ext_vector_type(8))) float v8f;

__global__ __launch_bounds__(WAVES_PER_BLOCK * 32)
void center_loss_partial(const float* __restrict__ x,
                         const int*   __restrict__ labels,
                         const float* __restrict__ centers,
                         float*       __restrict__ partial) {
    __shared__ float es[WAVES_PER_BLOCK][ROWS_PER_WAVE][LDS_STRIDE];

    const int lane      = threadIdx.x & 31;
    const int waveInBlk = threadIdx.x >> 5;
    const int wave      = blockIdx.x * WAVES_PER_BLOCK + waveInBlk;
    const int row0      = wave * ROWS_PER_WAVE;

    float (*E)[LDS_STRIDE] = es[waveInBlk];

    // ---- Stage E = X - gather(centers, labels) into wave-private LDS ----
    // One feature row per iteration: 32 lanes x float4 = 128 floats, coalesced.
    #pragma unroll 4
    for (int r = 0; r < ROWS_PER_WAVE; ++r) {
        const int    g   = row0 + r;
        const int    lbl = labels[g];                      // wave-uniform scalar load
        const float4 xv  = ((const float4*)(x + (size_t)g   * FEAT_DIM))[lane];
        const float4 cv  = ((const float4*)(centers + (size_t)lbl * FEAT_DIM))[lane];
        float4 ev;
        ev.x = xv.x - cv.x;  ev.y = xv.y - cv.y;
        ev.z = xv.z - cv.z;  ev.w = xv.w - cv.w;
        *(float4*)&E[r][lane * 4] = ev;
    }
    // LDS is wave-private and DS ops are in-order per wave: a dscnt drain is
    // sufficient for cross-lane visibility, no workgroup barrier needed.
    asm volatile("s_wait_dscnt 0x0" ::: "memory");

    // ---- Gram diagonal via V_WMMA_F32_16X16X4_F32, K accumulated 128 ----
    // A-fragment layout (16x4 f32): v0/v1 = K pair (phase,phase+1) for row M=lane%16,
    // phase = 2*(lane/16). B (4x16) of E^T uses the *same* registers.
    const int mrow  = lane & 15;
    const int phase = (lane >> 4) << 1;        // 0 for lanes 0-15, 2 for lanes 16-31

    v8f acc0 = {};
    v8f acc1 = {};
    #pragma unroll
    for (int t = 0; t < FEAT_DIM / 4; t += 2) {
        const float2 p0 = *(const float2*)&E[mrow][4 * t + phase];
        const float2 p1 = *(const float2*)&E[mrow][4 * (t + 1) + phase];
        v2f a0; a0.x = p0.x; a0.y = p0.y;
        v2f a1; a1.x = p1.x; a1.y = p1.y;
        acc0 = __builtin_amdgcn_wmma_f32_16x16x4_f32(false, a0, false, a0,
                                                     (short)0, acc0, false, false);
        acc1 = __builtin_amdgcn_wmma_f32_16x16x4_f32(false, a1, false, a1,
                                                     (short)0, acc1, false, false);
    }
    const v8f acc = acc0 + acc1;

    // ---- Extract diagonal D[m][m], clamp, wave-reduce ----
    // C/D layout: VGPR j holds M=j (lanes 0-15, N=lane) and M=j+8 (lanes 16-31,
    // N=lane-16).  Diag: m<8 -> lane m, vgpr m ; m>=8 -> lane m+16, vgpr m-8.
    const int s = (lane < 8) ? lane : ((lane >= 24) ? (lane - 24) : -1);
    float d = 0.0f;
    if (s >= 0) {
        float v = (s == 0) ? acc[0] : (s == 1) ? acc[1] : (s == 2) ? acc[2] :
                  (s == 3) ? acc[3] : (s == 4) ? acc[4] : (s == 5) ? acc[5] :
                  (s == 6) ? acc[6] : acc[7];
        d = fminf(fmaxf(v, 1e-12f), 1e12f);
    }
    #pragma unroll
    for (int off = 16; off > 0; off >>= 1)
        d += __shfl_xor(d, off, 32);
    if (lane == 0)
        partial[wave] = d;     // fully overwritten every call: deterministic
}

__global__ __launch_bounds__(32)
void center_loss_final(const float* __restrict__ partial,
                       float*       __restrict__ out) {
    const int lane = threadIdx.x;
    float s = partial[lane] + partial[lane + 32];       // NUM_WAVES == 64
    #pragma unroll
    for (int off = 16; off > 0; off >>= 1)
        s += __shfl_xor(s, off, 32);
    if (lane == 0) {
        // Every masked-off zero was clamped up to 1e-12: B*(C-1) of them.
        const float floor_term =
            (float)((double)BATCH * (double)(NUM_CLASSES - 1) * 1e-12);
        out[0] = (s + floor_term) / (float)BATCH;
    }
}

extern "C" void kernel_launch(void* const* d_in, const int* in_sizes, int n_in,
                              void* d_out, int out_size, void* d_ws, size_t ws_size,
                              hipStream_t stream) {
    const float* x       = (const float*)d_in[0];
    const int*   labels  = (const int*)d_in[1];   // jnp.int64 w/o x64 -> int32
    const float* centers = (const float*)d_in[2];
    float*       out     = (float*)d_out;
    float*       partial = (float*)d_ws;          // 64 floats of scratch

    const int blocks = BATCH / (ROWS_PER_WAVE * WAVES_PER_BLOCK);   // 16
    center_loss_partial<<<blocks, WAVES_PER_BLOCK * 32, 0, stream>>>(
        x, labels, centers, partial);
    center_loss_final<<<1, 32, 0, stream>>>(partial, out);
}